// myGINConv_40570261078603
// MI455X (gfx1250) — compile-verified
//
#include <hip/hip_runtime.h>
#include <cstddef>

typedef __attribute__((ext_vector_type(2))) float v2f;
typedef __attribute__((ext_vector_type(8))) float v8f;

#define DIM 64

// ---------------------------------------------------------------------------
// Kernel 1: zero the neigh accumulator (d_ws). Must run every launch since the
// harness does not re-poison/restore between graph replays.
// ---------------------------------------------------------------------------
__global__ void gin_zero_kernel(float* __restrict__ p, int n) {
  int i = blockIdx.x * blockDim.x + threadIdx.x;
  if (i < n) p[i] = 0.0f;
}

// ---------------------------------------------------------------------------
// Kernel 2: edge scatter. One wave32 per edge; lane l handles columns 2l,2l+1.
// feat (25.6MB) and neigh (25.6MB) both fit in the 192MB L2, so the random
// gather + f32 atomic-adds stay L2-resident.
// ---------------------------------------------------------------------------
__global__ void gin_scatter_kernel(const float* __restrict__ feat,
                                   const int* __restrict__ src,
                                   const int* __restrict__ dst,
                                   const float* __restrict__ ew,
                                   float* __restrict__ neigh, int E) {
  int gtid = blockIdx.x * blockDim.x + threadIdx.x;
  int e    = gtid >> 5;
  int lane = threadIdx.x & 31;
  if (e >= E) return;
  // All lanes read the same address -> single request; readfirstlane lets the
  // compiler keep the row bases in SGPRs.
  int   s = __builtin_amdgcn_readfirstlane(src[e]);
  int   d = __builtin_amdgcn_readfirstlane(dst[e]);
  float w = ew[e];

  const float2 v = *reinterpret_cast<const float2*>(feat + (size_t)s * DIM + lane * 2);
  float* nrow = neigh + (size_t)d * DIM + lane * 2;
  atomicAdd(nrow,     v.x * w);
  atomicAdd(nrow + 1, v.y * w);
}

// ---------------------------------------------------------------------------
// Kernel 3: fused (1+eps)*feat + neigh, x @ W^T + b, sigmoid — via
// V_WMMA_F32_16X16X4_F32 (full f32 precision, K=4 per issue, 16 steps).
// One wave per 16-node tile; 4 accumulators cover all 64 output columns.
//
// A frag (16x4, MxK):  lane -> (M = lane&15, K = 2*(lane>>4) + {0,1})
// B frag (4x16, KxN):  lane -> (N = lane&15, K = 2*(lane>>4) + {0,1})
//                      B[k][j] = W[j][k]  -> contiguous float2 in row-major W
// C/D (16x16):         vgpr v, lane -> (M = v + 8*(lane>>4), N = lane&15)
// ---------------------------------------------------------------------------
__global__ void gin_mlp_wmma_kernel(const float* __restrict__ feat,
                                    const float* __restrict__ neigh,
                                    const float* __restrict__ epsp,
                                    const float* __restrict__ Wm,
                                    const float* __restrict__ bias,
                                    float* __restrict__ out, int N) {
  const int lane = threadIdx.x & 31;
  const int tile = blockIdx.x * (blockDim.x >> 5) + (threadIdx.x >> 5);
  if (tile * 16 >= N) return;  // wave-uniform: EXEC stays all-1s for WMMA

  const int m    = lane & 15;
  const int kb   = (lane >> 4) << 1;          // 0 or 2
  const int row  = tile * 16 + m;
  const float e1 = 1.0f + epsp[0];

  const float* frow = feat  + (size_t)row * DIM;
  const float* nrow = neigh + (size_t)row * DIM;
  const int j = lane & 15;

  v8f acc0 = {}, acc1 = {}, acc2 = {}, acc3 = {};

#pragma unroll
  for (int k0 = 0; k0 < DIM; k0 += 4) {
    const int k = k0 + kb;
    const float2 f = *reinterpret_cast<const float2*>(frow + k);
    const float2 g = *reinterpret_cast<const float2*>(nrow + k);
    v2f a;
    a.x = fmaf(e1, f.x, g.x);
    a.y = fmaf(e1, f.y, g.y);

    const v2f b0 = *reinterpret_cast<const v2f*>(Wm + (size_t)(j     ) * DIM + k);
    const v2f b1 = *reinterpret_cast<const v2f*>(Wm + (size_t)(j + 16) * DIM + k);
    const v2f b2 = *reinterpret_cast<const v2f*>(Wm + (size_t)(j + 32) * DIM + k);
    const v2f b3 = *reinterpret_cast<const v2f*>(Wm + (size_t)(j + 48) * DIM + k);

    // (neg_a, A, neg_b, B, c_mod, C, reuse_a, reuse_b)
    acc0 = __builtin_amdgcn_wmma_f32_16x16x4_f32(false, a, false, b0, (short)0, acc0, false, false);
    acc1 = __builtin_amdgcn_wmma_f32_16x16x4_f32(false, a, false, b1, (short)0, acc1, false, false);
    acc2 = __builtin_amdgcn_wmma_f32_16x16x4_f32(false, a, false, b2, (short)0, acc2, false, false);
    acc3 = __builtin_amdgcn_wmma_f32_16x16x4_f32(false, a, false, b3, (short)0, acc3, false, false);
  }

  const int mbase = 8 * (lane >> 4);
  const int rbase = tile * 16 + mbase;

  auto emit = [&](const v8f& acc, int j0) {
    const float bj = bias[j0 + j];
#pragma unroll
    for (int v = 0; v < 8; ++v) {
      const float val = acc[v] + bj;
      out[(size_t)(rbase + v) * DIM + j0 + j] = 1.0f / (1.0f + __expf(-val));
    }
  };
  emit(acc0, 0);
  emit(acc1, 16);
  emit(acc2, 32);
  emit(acc3, 48);
}

// ---------------------------------------------------------------------------
// Launch. Inputs (setup_inputs order):
//  0 feat   [N,64] f32      3 edge_weight [E,1] f32   5 W [64,64] f32
//  1 src    [E]    i32      4 eps         [1]   f32   6 b [64]    f32
//  2 dst    [E]    i32
// d_out: [N,64] f32.  d_ws: neigh accumulator, N*64*4 = 25.6 MB.
// ---------------------------------------------------------------------------
extern "C" void kernel_launch(void* const* d_in, const int* in_sizes, int n_in,
                              void* d_out, int out_size, void* d_ws, size_t ws_size,
                              hipStream_t stream) {
  const float* feat = (const float*)d_in[0];
  const int*   src  = (const int*)d_in[1];
  const int*   dst  = (const int*)d_in[2];
  const float* ew   = (const float*)d_in[3];
  const float* eps  = (const float*)d_in[4];
  const float* Wm   = (const float*)d_in[5];
  const float* bias = (const float*)d_in[6];
  float*       out  = (float*)d_out;
  float*       neigh = (float*)d_ws;

  const int N = in_sizes[0] / DIM;   // 100000
  const int E = in_sizes[1];         // 1600000

  // 1) zero neigh
  const int nElems = N * DIM;
  gin_zero_kernel<<<(nElems + 255) / 256, 256, 0, stream>>>(neigh, nElems);

  // 2) scatter: one wave per edge -> 8 edges per 256-thread block
  const int scatterBlocks = (E + 7) / 8;
  gin_scatter_kernel<<<scatterBlocks, 256, 0, stream>>>(feat, src, dst, ew, neigh, E);

  // 3) WMMA MLP: one wave per 16-row tile, 8 waves per block
  const int tiles = (N + 15) / 16;
  gin_mlp_wmma_kernel<<<(tiles + 7) / 8, 256, 0, stream>>>(feat, neigh, eps, Wm, bias, out, N);
}